// Net_80582176408220
// MI455X (gfx1250) — compile-verified
//
#include <hip/hip_runtime.h>
#include <hip/hip_bf16.h>
#include <math.h>

typedef __attribute__((ext_vector_type(2))) float v2f;
typedef __attribute__((ext_vector_type(8))) float v8f;

#define H 768
#define B 64
#define S 512            // sequence length after dropping CLS token
#define T 4              // NLAB + 2
#define START 2
#define STOP 3

// ---------------------------------------------------------------------------
// Kernel 1: feats[b,s,t] = emb[b, s+1, :] . crf_W[t, :] + crf_b[t]
// GEMM (B*S = 32768 rows) x (K=768) x (N=4, zero-padded to 16) via
// V_WMMA_F32_16X16X4_F32. Each wave owns FOUR 16-row M-tiles (64 rows) so the
// shared B ds_load is amortized over 4 independent WMMAs per K-step.
// B operand lives zero-padded in LDS -> unconditional ds_load_b64 (no exec
// masking in the hot loop).
// ---------------------------------------------------------------------------
__global__ void __launch_bounds__(256)
feats_wmma_kernel(const float* __restrict__ emb,
                  const float* __restrict__ crf_W,
                  const float* __restrict__ crf_b,
                  float* __restrict__ feats) {
    __shared__ float sWp[16 * H];               // padded W, [n][k], 48 KB
    const int tid = threadIdx.x;
    for (int i = tid; i < 16 * H; i += 256) {
        const int n = i / H;
        const int k = i % H;
        sWp[i] = (n < T) ? crf_W[n * H + k] : 0.0f;
    }
    __syncthreads();

    const int wave  = tid >> 5;
    const int lane  = tid & 31;
    const int group = blockIdx.x * 8 + wave;    // 0..511, 4 tiles each
    const int m0    = group * 64;               // flat row = b*S + s
    const int b     = m0 / S;
    const int s0    = m0 % S;                   // 512 % 64 == 0 -> no batch crossing

    // emb row for flat row m: b*513 + (s+1)
    const float* Abase = emb + ((size_t)b * (S + 1) + s0 + 1) * H;

    const int row   = lane & 15;                // M within tile (A) / N (B)
    const int khalf = (lane >> 4) * 2;          // lanes 0-15: K={0,1}; 16-31: K={2,3}

    const float* blds = sWp + (size_t)row * H + khalf;   // per-lane B pointer
    const float* a0p  = Abase + (size_t)row * H + khalf; // tile 0 A pointer

    v8f acc0 = {}, acc1 = {}, acc2 = {}, acc3 = {};
    for (int k0 = 0; k0 < H; k0 += 4) {
        const v2f bv = *(const v2f*)(blds + k0);                 // ds_load_b64
        const v2f A0 = *(const v2f*)(a0p + k0);                  // global_load_b64
        const v2f A1 = *(const v2f*)(a0p + k0 + 16 * H);
        const v2f A2 = *(const v2f*)(a0p + k0 + 32 * H);
        const v2f A3 = *(const v2f*)(a0p + k0 + 48 * H);
        acc0 = __builtin_amdgcn_wmma_f32_16x16x4_f32(false, A0, false, bv,
                                                     (short)0, acc0, false, false);
        acc1 = __builtin_amdgcn_wmma_f32_16x16x4_f32(false, A1, false, bv,
                                                     (short)0, acc1, false, false);
        acc2 = __builtin_amdgcn_wmma_f32_16x16x4_f32(false, A2, false, bv,
                                                     (short)0, acc2, false, false);
        acc3 = __builtin_amdgcn_wmma_f32_16x16x4_f32(false, A3, false, bv,
                                                     (short)0, acc3, false, false);
    }

    // D layout: lanes 0-15 hold N=lane, M = vgpr index i (0..7);
    //           lanes 16-31 hold N=lane-16, M = i + 8.
    const int col   = lane & 15;
    const int mbase = (lane < 16) ? 0 : 8;
    if (col < T) {
        const float bias = crf_b[col];
        float* d0 = feats + (size_t)(m0 + mbase) * T + col;
#pragma unroll
        for (int i = 0; i < 8; ++i) d0[(size_t)i * T]            = acc0[i] + bias;
#pragma unroll
        for (int i = 0; i < 8; ++i) d0[(size_t)(16 + i) * T]     = acc1[i] + bias;
#pragma unroll
        for (int i = 0; i < 8; ++i) d0[(size_t)(32 + i) * T]     = acc2[i] + bias;
#pragma unroll
        for (int i = 0; i < 8; ++i) d0[(size_t)(48 + i) * T]     = acc3[i] + bias;
    }
}

// ---------------------------------------------------------------------------
// Kernel 2: isqa head. One block per batch row.
// ---------------------------------------------------------------------------
__global__ void __launch_bounds__(256)
isqa_kernel(const float* __restrict__ emb,
            const float* __restrict__ fc2_W,
            const float* __restrict__ fc2_b,
            const int* __restrict__ IsQA,
            float* __restrict__ out_pred,    // d_out + 0        (64)
            float* __restrict__ out_isqa,    // d_out + 65602    (64)
            float* __restrict__ qb) {        // per-batch nll
    const int b = blockIdx.x;
    __shared__ float s0[256], s1[256];
    const float* x = emb + (size_t)b * (S + 1) * H;   // CLS row (s == 0)
    float a0 = 0.f, a1 = 0.f;
    for (int h = threadIdx.x; h < H; h += 256) {
        const float v = x[h];
        a0 += v * fc2_W[h];
        a1 += v * fc2_W[H + h];
    }
    s0[threadIdx.x] = a0;
    s1[threadIdx.x] = a1;
    __syncthreads();
    for (int off = 128; off > 0; off >>= 1) {
        if (threadIdx.x < off) {
            s0[threadIdx.x] += s0[threadIdx.x + off];
            s1[threadIdx.x] += s1[threadIdx.x + off];
        }
        __syncthreads();
    }
    if (threadIdx.x == 0) {
        const float l0 = s0[0] + fc2_b[0];
        const float l1 = s1[0] + fc2_b[1];
        const int pred = (l1 > l0) ? 1 : 0;           // argmax, first-index tiebreak
        const float mx = fmaxf(l0, l1);
        const float lse = mx + logf(expf(l0 - mx) + expf(l1 - mx));
        const int y = IsQA[b];
        const float logp = ((y == 0) ? l0 : l1) - lse;
        qb[b]       = -logp;
        out_pred[b] = (float)pred;
        out_isqa[b] = (float)y;
    }
}

// ---------------------------------------------------------------------------
// Kernel 3: fused CRF log-partition + gold score + Viterbi (+ backtrack).
// One thread per batch element; T=4 states live in registers.
// ---------------------------------------------------------------------------
__global__ void __launch_bounds__(64)
crf_kernel(const float* __restrict__ feats,
           const int* __restrict__ labels,        // answer_seq_label (B, S+1)
           const float* __restrict__ trans,       // (T, T)
           unsigned char* __restrict__ bp,        // ws: B*S*T bytes
           float* __restrict__ zb,                // per-batch (Z - gold)
           float* __restrict__ crf_pred_out,      // d_out + 64   (B*S)
           float* __restrict__ tags_out) {        // d_out + 32834 (B*S)
    const int b = threadIdx.x;
    if (b >= B) return;

    float tr[T][T];
#pragma unroll
    for (int i = 0; i < T; ++i)
#pragma unroll
        for (int j = 0; j < T; ++j) tr[i][j] = trans[i * T + j];

    const float* f = feats + (size_t)b * S * T;
    const int* tg = labels + (size_t)b * (S + 1) + 1;   // tags[b, s] = labels[b, s+1]
    unsigned char* bpb = bp + (size_t)b * S * T;

    float alpha[T], delta[T];
#pragma unroll
    for (int j = 0; j < T; ++j) {
        const float v = f[j] + tr[START][j];
        alpha[j] = v;
        delta[j] = v;
    }

    int prev = tg[0];
    float gold = f[prev] + tr[START][prev];

    for (int s = 1; s < S; ++s) {
        const float* ft = f + s * T;
        float na[T], nd[T];
#pragma unroll
        for (int j = 0; j < T; ++j) {
            // logsumexp_i(alpha[i] + tr[i][j])
            float m = alpha[0] + tr[0][j];
#pragma unroll
            for (int i = 1; i < T; ++i) m = fmaxf(m, alpha[i] + tr[i][j]);
            float ssum = 0.f;
#pragma unroll
            for (int i = 0; i < T; ++i) ssum += expf(alpha[i] + tr[i][j] - m);
            na[j] = m + logf(ssum) + ft[j];
            // Viterbi max + backpointer (first-index tiebreak)
            float best = delta[0] + tr[0][j];
            int bi = 0;
#pragma unroll
            for (int i = 1; i < T; ++i) {
                const float v = delta[i] + tr[i][j];
                if (v > best) { best = v; bi = i; }
            }
            nd[j] = best + ft[j];
            bpb[s * T + j] = (unsigned char)bi;
        }
#pragma unroll
        for (int j = 0; j < T; ++j) { alpha[j] = na[j]; delta[j] = nd[j]; }
        const int t = tg[s];
        gold += ft[t] + tr[prev][t];
        prev = t;
    }
    gold += tr[prev][STOP];

    // Z = logsumexp(alpha + tr[:,STOP])
    float m = alpha[0] + tr[0][STOP];
#pragma unroll
    for (int i = 1; i < T; ++i) m = fmaxf(m, alpha[i] + tr[i][STOP]);
    float ssum = 0.f;
#pragma unroll
    for (int i = 0; i < T; ++i) ssum += expf(alpha[i] + tr[i][STOP] - m);
    zb[b] = (m + logf(ssum)) - gold;

    // last tag = argmax(delta + tr[:,STOP])
    float best = delta[0] + tr[0][STOP];
    int lt = 0;
#pragma unroll
    for (int i = 1; i < T; ++i) {
        const float v = delta[i] + tr[i][STOP];
        if (v > best) { best = v; lt = i; }
    }

    // backtrack
    float* pred = crf_pred_out + (size_t)b * S;
    pred[S - 1] = (float)lt;
    int cur = lt;
    for (int s = S - 1; s >= 1; --s) {
        cur = bpb[s * T + cur];
        pred[s - 1] = (float)cur;
    }

    // tags output (int -> float)
    for (int s = 0; s < S; ++s) tags_out[(size_t)b * S + s] = (float)tg[s];
}

// ---------------------------------------------------------------------------
// Kernel 4: reduce per-batch losses to the two scalar output slots.
// ---------------------------------------------------------------------------
__global__ void __launch_bounds__(64)
finalize_kernel(const float* __restrict__ qb,
                const float* __restrict__ zb,
                float* __restrict__ out_losses) {   // [0]=isqa_loss, [1]=crf_loss
    __shared__ float sq[B], sz[B];
    const int t = threadIdx.x;
    sq[t] = qb[t];
    sz[t] = zb[t];
    __syncthreads();
    for (int off = 32; off > 0; off >>= 1) {
        if (t < off) { sq[t] += sq[t + off]; sz[t] += sz[t + off]; }
        __syncthreads();
    }
    if (t == 0) {
        out_losses[0] = sq[0] / (float)B;
        out_losses[1] = sz[0] / (float)B;
    }
}

extern "C" void kernel_launch(void* const* d_in, const int* in_sizes, int n_in,
                              void* d_out, int out_size, void* d_ws, size_t ws_size,
                              hipStream_t stream) {
    (void)in_sizes; (void)n_in; (void)out_size; (void)ws_size;

    const float* emb    = (const float*)d_in[0];   // (64, 513, 768)
    const int*   labels = (const int*)d_in[1];     // (64, 513)
    const int*   IsQA   = (const int*)d_in[2];     // (64,)
    const float* fc2_W  = (const float*)d_in[3];   // (2, 768)
    const float* fc2_b  = (const float*)d_in[4];   // (2,)
    const float* crf_W  = (const float*)d_in[5];   // (4, 768)
    const float* crf_b  = (const float*)d_in[6];   // (4,)
    const float* trans  = (const float*)d_in[7];   // (4, 4)

    float* out = (float*)d_out;
    // Output layout (flat floats, return order):
    float* out_isqa_pred = out;                      // 64
    float* out_crf_pred  = out + 64;                 // 32768
    float* out_losses    = out + 64 + B * S;         // [32832]=isqa, [32833]=crf
    float* out_tags      = out + 64 + B * S + 2;     // 32768
    float* out_IsQA      = out + 64 + 2 * B * S + 2; // 64

    // Workspace layout:
    float* feats = (float*)d_ws;                               // B*S*T floats (512 KB)
    float* zb    = feats + (size_t)B * S * T;                  // 64 floats
    float* qb    = zb + B;                                     // 64 floats
    unsigned char* bp = (unsigned char*)(qb + B);              // B*S*T bytes (128 KB)

    // 1) feats GEMM via WMMA: 512 wave-groups (4 tiles each), 8 waves/block.
    feats_wmma_kernel<<<64, 256, 0, stream>>>(emb, crf_W, crf_b, feats);

    // 2) isqa head.
    isqa_kernel<<<B, 256, 0, stream>>>(emb, fc2_W, fc2_b, IsQA,
                                       out_isqa_pred, out_IsQA, qb);

    // 3) fused CRF scans + backtrack.
    crf_kernel<<<1, B, 0, stream>>>(feats, labels, trans, bp, zb,
                                    out_crf_pred, out_tags);

    // 4) scalar losses.
    finalize_kernel<<<1, B, 0, stream>>>(qb, zb, out_losses);
}